// AdaptiveDistillationLoss_41034117546381
// MI455X (gfx1250) — compile-verified
//
#include <hip/hip_runtime.h>
#include <hip/hip_bf16.h>
#include <cstdint>

// ---------------------------------------------------------------------------
// AdaptiveDistillationLoss for MI455X (gfx1250)
// HBM-bound streaming: 1 GB of f32 reads -> ~44us floor at 23.3 TB/s.
//  - logits: CDNA5 Tensor Data Mover double-buffers 16 KB row tiles into LDS
//    (tensor_load_to_lds + s_wait_tensorcnt), consumed by 16 wave32s.
//  - soft_labels: streamed straight to VGPRs with nontemporal b128 loads
//    (consumed once; NT hint keeps the 192 MB L2 for the logits stream).
//  - single-pass online log-sum-exp => logits are read exactly once.
// ---------------------------------------------------------------------------

typedef uint32_t v4u __attribute__((ext_vector_type(4)));
typedef int      v8i __attribute__((ext_vector_type(8)));
typedef int      v4i __attribute__((ext_vector_type(4)));
typedef float    v4f __attribute__((ext_vector_type(4)));

#define LOG2E 1.44269504088896340736f
#define LN2   0.69314718055994530942f

#define BDIM   512      // 16 wave32s
#define CHUNK  4096     // f32 elements per TDM tile (16 KB); 32000 = 7 full + tail

// Issue one 1-D TDM load: tile of `tile_elems` f32 starting at gaddr -> LDS.
// tensor_dim0 = elements remaining in the row => OOB tail is zero-filled.
__device__ __forceinline__ void tdm_load_1d(uint32_t lds_byte_off,
                                            uint64_t gaddr,
                                            uint32_t remaining_elems,
                                            uint32_t tile_elems) {
  v4u g0;
  g0.x = 1u;                                              // count=1, user D#
  g0.y = lds_byte_off;                                    // D#.lds_addr
  g0.z = (uint32_t)(gaddr & 0xFFFFFFFFull);               // global_addr[31:0]
  g0.w = (uint32_t)((gaddr >> 32) & 0x1FFFFFFull)         // global_addr[56:32]
         | (2u << 30);                                    // type = 2 ("image")
  v8i g1;
  g1[0] = (int)(2u << 16);                                // data_size=2 (4B), wg_mask=0
  g1[1] = (int)((remaining_elems & 0xFFFFu) << 16);       // tensor_dim0[15:0]
  g1[2] = (int)(((remaining_elems >> 16) & 0xFFFFu)       // tensor_dim0[31:16]
         | (1u << 16));                                   // tensor_dim1 = 1
  g1[3] = (int)((tile_elems & 0xFFFFu) << 16);            // tile_dim0
  g1[4] = 0;                                              // tile_dim1/2 unused
  g1[5] = (int)remaining_elems;                           // dim0 stride (unused, 1D)
  g1[6] = 0;
  g1[7] = 0;
  v4i z4 = {0, 0, 0, 0};                                  // groups 2/3: <=2D tensor
#if __has_include(<hip/amd_detail/amd_gfx1250_TDM.h>)
  v8i z8 = {0, 0, 0, 0, 0, 0, 0, 0};
  __builtin_amdgcn_tensor_load_to_lds(g0, g1, z4, z4, z8, 0);
#else
  __builtin_amdgcn_tensor_load_to_lds(g0, g1, z4, z4, 0);
#endif
}

// Flat LDS pointer keeps the wave-relative LDS byte offset in its low 32 bits
// (ISA 10.2: LDS aperture address = {SHARED_BASE, offset32}).
__device__ __forceinline__ uint32_t lds_off32(const void* p) {
  return (uint32_t)(uintptr_t)p;
}

__global__ __launch_bounds__(BDIM)
void adl_row_kernel(const float* __restrict__ logits,
                    const int*   __restrict__ labels,
                    const float* __restrict__ soft,
                    const float* __restrict__ conf,
                    float* __restrict__ ce_rows,
                    float* __restrict__ kl_rows,
                    float* __restrict__ t_rows,
                    int C) {
  __shared__ float sh_lg[2][CHUNK];
  __shared__ float red[BDIM / 32][5];

  const int r   = blockIdx.x;
  const int tid = threadIdx.x;

  // adaptive temperature (per-row scalar)
  const float cf   = conf[r];
  const float low  = fminf(2.5f + (0.6f - cf) * 2.0f, 3.0f);
  const float T    = (cf > 0.9f) ? 1.5f : ((cf > 0.6f) ? 2.0f : low);
  const float invT = 1.0f / T;

  const uint64_t rowL = (uint64_t)(uintptr_t)logits + (uint64_t)r * (uint64_t)C * 4ull;
  const v4f* __restrict__ sf4 = (const v4f*)(soft + (uint64_t)r * (uint64_t)C);
  const int nch = (C + CHUNK - 1) / CHUNK;

  // Prime the pipeline: wave 0 issues chunk 0 (TDM ignores EXEC, so this must
  // be a wave-uniform region that other waves branch around entirely).
  if (tid < 32) {
    tdm_load_1d(lds_off32(&sh_lg[0][0]), rowL, (uint32_t)C, CHUNK);
  }

  // online-softmax state
  float m = -3.0e38f, s1 = 0.0f, sT = 0.0f, stx = 0.0f, stl = 0.0f;

  for (int c = 0; c < nch; ++c) {
    const int buf = c & 1;
    if (tid < 32) {
      if (c + 1 < nch) {  // prefetch next chunk into the other buffer
        const uint64_t off = (uint64_t)(c + 1) * (uint64_t)CHUNK * 4ull;
        const uint32_t rem = (uint32_t)(C - (c + 1) * CHUNK);
        tdm_load_1d(lds_off32(&sh_lg[buf ^ 1][0]), rowL + off, rem, CHUNK);
        __builtin_amdgcn_s_wait_tensorcnt(1);   // chunk c landed (in-order)
      } else {
        __builtin_amdgcn_s_wait_tensorcnt(0);
      }
    }
    __syncthreads();  // publish chunk c to all 16 waves

    const int len  = min(CHUNK, C - c * CHUNK);
    const int len4 = len >> 2;
    const int g0_4 = c * (CHUNK >> 2);            // row-global float4 index base
    const v4f* lg4 = (const v4f*)&sh_lg[buf][0];
    for (int j = tid; j < len4; j += BDIM) {
      const v4f x4 = lg4[j];                                     // LDS (TDM-fed)
      const v4f t4 = __builtin_nontemporal_load(&sf4[g0_4 + j]); // HBM stream
#pragma unroll
      for (int k = 0; k < 4; ++k) {
        const float x = x4[k];
        const float t = t4[k];
        if (x > m) {                       // rare: rescale running sums
          const float d = (m - x) * LOG2E;
          s1 = s1 * exp2f(d) + 1.0f;
          sT = sT * exp2f(d * invT) + 1.0f;
          m = x;
        } else {
          const float d = (x - m) * LOG2E;
          s1 += exp2f(d);
          sT += exp2f(d * invT);
        }
        stx += t * x;
        stl += (t > 0.0f) ? t * __log2f(t) : 0.0f;   // xlogy, in log2 units
      }
    }
    __syncthreads();  // done reading buf before TDM may overwrite it
  }

  // ---- wave32 reduction: merge (m, s1, sT) with log-sum-exp combine ----
#pragma unroll
  for (int off = 16; off > 0; off >>= 1) {
    const float mo = __shfl_xor(m,  off, 32);
    const float so = __shfl_xor(s1, off, 32);
    const float to = __shfl_xor(sT, off, 32);
    const float M  = fmaxf(m, mo);
    s1 = s1 * exp2f((m - M) * LOG2E)        + so * exp2f((mo - M) * LOG2E);
    sT = sT * exp2f((m - M) * LOG2E * invT) + to * exp2f((mo - M) * LOG2E * invT);
    m  = M;
    stx += __shfl_xor(stx, off, 32);
    stl += __shfl_xor(stl, off, 32);
  }
  const int wave = tid >> 5, lane = tid & 31;
  if (lane == 0) {
    red[wave][0] = m;  red[wave][1] = s1; red[wave][2] = sT;
    red[wave][3] = stx; red[wave][4] = stl;
  }
  __syncthreads();
  if (wave == 0) {
    const int nw = BDIM / 32;  // 16
    m   = (lane < nw) ? red[lane][0] : -3.0e38f;
    s1  = (lane < nw) ? red[lane][1] : 0.0f;
    sT  = (lane < nw) ? red[lane][2] : 0.0f;
    stx = (lane < nw) ? red[lane][3] : 0.0f;
    stl = (lane < nw) ? red[lane][4] : 0.0f;
#pragma unroll
    for (int off = 16; off > 0; off >>= 1) {
      const float mo = __shfl_xor(m,  off, 32);
      const float so = __shfl_xor(s1, off, 32);
      const float to = __shfl_xor(sT, off, 32);
      const float M  = fmaxf(m, mo);
      s1 = s1 * exp2f((m - M) * LOG2E)        + so * exp2f((mo - M) * LOG2E);
      sT = sT * exp2f((m - M) * LOG2E * invT) + to * exp2f((mo - M) * LOG2E * invT);
      m  = M;
      stx += __shfl_xor(stx, off, 32);
      stl += __shfl_xor(stl, off, 32);
    }
    if (lane == 0) {
      const int   lab  = labels[r];
      const float xlab = logits[(uint64_t)r * (uint64_t)C + (uint64_t)lab];
      const float lnZ1 = __log2f(s1) * LN2;
      const float lnZT = __log2f(sT) * LN2;
      ce_rows[r] = m + lnZ1 - xlab;                          // -logp[label]
      kl_rows[r] = stl * LN2 - stx * invT + m * invT + lnZT; // KL(t || softmax(x/T))
      t_rows[r]  = T;
    }
  }
}

__global__ __launch_bounds__(1024)
void adl_finalize_kernel(const float* __restrict__ ce_rows,
                         const float* __restrict__ kl_rows,
                         const float* __restrict__ t_rows,
                         float* __restrict__ out, int B) {
  __shared__ float sc[32], sk[32], st[32];
  float a = 0.0f, b = 0.0f, c = 0.0f;
  for (int i = threadIdx.x; i < B; i += 1024) {
    a += ce_rows[i]; b += kl_rows[i]; c += t_rows[i];
  }
#pragma unroll
  for (int off = 16; off > 0; off >>= 1) {
    a += __shfl_xor(a, off, 32);
    b += __shfl_xor(b, off, 32);
    c += __shfl_xor(c, off, 32);
  }
  const int wave = threadIdx.x >> 5, lane = threadIdx.x & 31;
  if (lane == 0) { sc[wave] = a; sk[wave] = b; st[wave] = c; }
  __syncthreads();
  if (wave == 0) {
    a = sc[lane]; b = sk[lane]; c = st[lane];  // 1024 threads = 32 waves exactly
#pragma unroll
    for (int off = 16; off > 0; off >>= 1) {
      a += __shfl_xor(a, off, 32);
      b += __shfl_xor(b, off, 32);
      c += __shfl_xor(c, off, 32);
    }
    if (lane == 0) {
      const float invB = 1.0f / (float)B;
      const float ce = a * invB, kl = b * invB, at = c * invB;
      out[0] = 0.5f * kl + 0.5f * ce;  // total (ALPHA = 0.5)
      out[1] = ce;
      out[2] = kl;
      out[3] = at;
    }
  }
}

extern "C" void kernel_launch(void* const* d_in, const int* in_sizes, int n_in,
                              void* d_out, int out_size, void* d_ws, size_t ws_size,
                              hipStream_t stream) {
  (void)n_in; (void)out_size; (void)ws_size;
  const float* logits = (const float*)d_in[0];
  const int*   labels = (const int*)d_in[1];
  const float* soft   = (const float*)d_in[2];
  const float* conf   = (const float*)d_in[3];
  const int B = in_sizes[1];             // 4096
  const int C = in_sizes[0] / B;         // 32000

  float* ws      = (float*)d_ws;         // 3*B floats of scratch (48 KB)
  float* ce_rows = ws;
  float* kl_rows = ws + B;
  float* t_rows  = ws + 2 * B;

  adl_row_kernel<<<B, BDIM, 0, stream>>>(logits, labels, soft, conf,
                                         ce_rows, kl_rows, t_rows, C);
  adl_finalize_kernel<<<1, 1024, 0, stream>>>(ce_rows, kl_rows, t_rows,
                                              (float*)d_out, B);
}